// CorefSeq_12850542150166
// MI455X (gfx1250) — compile-verified
//
#include <hip/hip_runtime.h>

// Problem constants (from the reference)
#define B_   1024
#define S_   512
#define H_   768
#define H3_  2304   // 3*H
#define HID_ 512
#define NC_  3

typedef float v2f __attribute__((ext_vector_type(2)));
typedef float v8f __attribute__((ext_vector_type(8)));

// ---------------------------------------------------------------------------
// Kernel 1: three masked segment-means over the sequence dimension.
// One block per batch row. Labels staged in LDS so the per-s branch is
// block-uniform (no divergence). Each of the 192 threads owns one float4
// column group (192*4 = 768 = H) -> perfectly coalesced 16B loads, streaming
// the 1.6 GB embeddings tensor exactly once (the HBM roofline term).
// ---------------------------------------------------------------------------
__global__ __launch_bounds__(192) void pool_kernel(
    const float* __restrict__ emb,      // [B, S, H]
    const int*   __restrict__ mentions, // [B, S]
    float*       __restrict__ pooled)   // [B, 3H]
{
  __shared__ int labs[S_];
  __shared__ int cnt[3];

  const int b   = blockIdx.x;
  const int tid = threadIdx.x;

  if (tid < 3) cnt[tid] = 0;
  __syncthreads();

  int c2 = 0, c3 = 0, c4 = 0;
  for (int s = tid; s < S_; s += 192) {
    int lab = mentions[b * S_ + s];
    labs[s] = lab;
    c2 += (lab == 2);
    c3 += (lab == 3);
    c4 += (lab == 4);
  }
  if (c2) atomicAdd(&cnt[0], c2);
  if (c3) atomicAdd(&cnt[1], c3);
  if (c4) atomicAdd(&cnt[2], c4);
  __syncthreads();

  const float inv2 = 1.0f / (float)cnt[0];
  const float inv3 = 1.0f / (float)cnt[1];
  const float inv4 = 1.0f / (float)cnt[2];

  // Each thread accumulates one float4 column group across all S rows.
  const float4* src = (const float4*)(emb + (size_t)b * S_ * H_) + tid;
  float a2x = 0.f, a2y = 0.f, a2z = 0.f, a2w = 0.f;
  float a3x = 0.f, a3y = 0.f, a3z = 0.f, a3w = 0.f;
  float a4x = 0.f, a4y = 0.f, a4z = 0.f, a4w = 0.f;

#pragma unroll 4
  for (int s = 0; s < S_; ++s) {
    float4 v = src[(size_t)s * (H_ / 4)];
    int lab = labs[s];               // block-uniform
    if (lab == 2) {
      a2x += v.x; a2y += v.y; a2z += v.z; a2w += v.w;
    } else if (lab == 3) {
      a3x += v.x; a3y += v.y; a3z += v.z; a3w += v.w;
    } else if (lab == 4) {
      a4x += v.x; a4y += v.y; a4z += v.z; a4w += v.w;
    }
  }

  float4* dst = (float4*)(pooled + (size_t)b * H3_);
  dst[tid]                = make_float4(a2x * inv2, a2y * inv2, a2z * inv2, a2w * inv2);
  dst[tid + (H_ / 4)]     = make_float4(a3x * inv3, a3y * inv3, a3z * inv3, a3w * inv3);
  dst[tid + 2 * (H_ / 4)] = make_float4(a4x * inv4, a4y * inv4, a4z * inv4, a4w * inv4);
}

// ---------------------------------------------------------------------------
// Kernel 2: h = relu(pooled @ W1 + b1) via V_WMMA_F32_16X16X4_F32.
// One wave computes one 16x16 tile of h[1024,512]; 4 waves per block cover
// 4 adjacent N-tiles. K loop of 2304 in steps of 4.
//
// fp32 WMMA fragment layout (ISA 7.12.2):
//   A (16x4): lanes 0-15 -> rows M=lane, K = {0,1}; lanes 16-31 -> K = {2,3}
//   B (4x16): lanes 0-15 -> cols N=lane, K = {0,1}; lanes 16-31 -> K = {2,3}
//   C/D (16x16): VGPR r, lane l -> row M = r + 8*(l>=16), col N = l&15
// ---------------------------------------------------------------------------
__global__ __launch_bounds__(128) void mlp1_kernel(
    const float* __restrict__ pooled, // [B, 3H]
    const float* __restrict__ W1,     // [3H, 512] row-major (K x N)
    const float* __restrict__ b1,     // [512]
    float*       __restrict__ h)      // [B, 512]
{
  const int lane  = threadIdx.x & 31;
  const int wave  = threadIdx.x >> 5;
  const int m0    = blockIdx.y * 16;
  const int n0    = (blockIdx.x * 4 + wave) * 16;
  const int mrow  = lane & 15;       // A row within tile
  const int ncol  = lane & 15;       // B/C column within tile
  const int khalf = (lane >> 4) * 2; // 0 for lanes 0-15, 2 for lanes 16-31

  v8f acc = {0.f, 0.f, 0.f, 0.f, 0.f, 0.f, 0.f, 0.f};

  const float* aptr = pooled + (size_t)(m0 + mrow) * H3_ + khalf;
  const float* bptr = W1 + (size_t)khalf * HID_ + n0 + ncol;

#pragma unroll 4
  for (int k = 0; k < H3_; k += 4) {
    v2f a = *(const v2f*)(aptr + k);             // 8B-aligned contiguous pair
    v2f bb;
    bb.x = bptr[(size_t)k * HID_];
    bb.y = bptr[(size_t)(k + 1) * HID_];
    acc = __builtin_amdgcn_wmma_f32_16x16x4_f32(
        /*neg_a=*/false, a, /*neg_b=*/false, bb,
        /*c_mod=*/(short)0, acc, /*reuse_a=*/false, /*reuse_b=*/false);
  }

  const float bias  = b1[n0 + ncol];
  const int   rbase = (lane >> 4) * 8;
#pragma unroll
  for (int r = 0; r < 8; ++r) {
    float v = acc[r] + bias;
    v = v > 0.f ? v : 0.f;
    h[(size_t)(m0 + rbase + r) * HID_ + n0 + ncol] = v;
  }
}

// ---------------------------------------------------------------------------
// Kernel 3: out = h @ W2 + b2   (512 x 3 head; trivial reduction per row)
// ---------------------------------------------------------------------------
__global__ __launch_bounds__(128) void mlp2_kernel(
    const float* __restrict__ h,   // [B, 512]
    const float* __restrict__ W2,  // [512, 3]
    const float* __restrict__ b2,  // [3]
    float*       __restrict__ out) // [B, 3]
{
  __shared__ float red[3][128];
  const int b   = blockIdx.x;
  const int tid = threadIdx.x;

  float p0 = 0.f, p1 = 0.f, p2 = 0.f;
#pragma unroll
  for (int k = tid; k < HID_; k += 128) {
    float hv = h[(size_t)b * HID_ + k];
    p0 += hv * W2[k * 3 + 0];
    p1 += hv * W2[k * 3 + 1];
    p2 += hv * W2[k * 3 + 2];
  }
  red[0][tid] = p0; red[1][tid] = p1; red[2][tid] = p2;
  __syncthreads();
  for (int off = 64; off > 0; off >>= 1) {
    if (tid < off) {
      red[0][tid] += red[0][tid + off];
      red[1][tid] += red[1][tid + off];
      red[2][tid] += red[2][tid + off];
    }
    __syncthreads();
  }
  if (tid < 3) out[b * 3 + tid] = red[tid][0] + b2[tid];
}

// ---------------------------------------------------------------------------
extern "C" void kernel_launch(void* const* d_in, const int* in_sizes, int n_in,
                              void* d_out, int out_size, void* d_ws, size_t ws_size,
                              hipStream_t stream) {
  const float* emb      = (const float*)d_in[0]; // [B,S,H]
  const int*   mentions = (const int*)d_in[1];   // [B,S]
  const float* W1       = (const float*)d_in[2]; // [3H,512]
  const float* b1       = (const float*)d_in[3]; // [512]
  const float* W2       = (const float*)d_in[4]; // [512,3]
  const float* b2       = (const float*)d_in[5]; // [3]
  float*       out      = (float*)d_out;         // [B,3]

  // Workspace layout: pooled [B,3H] fp32, then h [B,512] fp32  (~11.5 MB)
  float* pooled = (float*)d_ws;
  float* h      = pooled + (size_t)B_ * H3_;

  pool_kernel<<<B_, 192, 0, stream>>>(emb, mentions, pooled);
  mlp1_kernel<<<dim3(HID_ / 64, B_ / 16), 128, 0, stream>>>(pooled, W1, b1, h);
  mlp2_kernel<<<B_, 128, 0, stream>>>(h, W2, b2, out);
}